// ProductVectorQuantize_43241730736411
// MI455X (gfx1250) — compile-verified
//
#include <hip/hip_runtime.h>
#include <hip/hip_bf16.h>
#include <math.h>

typedef __attribute__((ext_vector_type(2))) float v2f;
typedef __attribute__((ext_vector_type(8))) float v8f;

#define NVQ   6
#define CDIM  32
#define KCODE 1024
#define HH    5
#define CC    256
#define OVV   4
#define WT    1200
#define TOV   300            // Wt / OV
#define NROWS 9600           // B * TOV
#define FIXD  1280           // C * H
#define KPAD  864            // 853/855 padded to 8 waves * 27 steps * 4
#define TSTR  866            // LDS tile stride (bank-conflict-free: 866 mod 64 = 34)
#define NW    8              // waves per block (256 threads, wave32)

// output layout (floats), concatenated flat in return order
#define ZQ_N    49152000LL   // 32*6000*256
#define ZD_OFF  49152000LL
#define IDX_OFF 50995200LL   // + 32*6*300*32
#define CM_OFF  51052800LL   // + 32*6*300

struct Params {
  const float* z_e;
  const float* Wd[NVQ]; const float* bd[NVQ];
  const float* Wu[NVQ]; const float* bu[NVQ];
  const float* emb[NVQ];
  float* out;
  float* cbrow;   // (6,1024,32) normalized codebook
  float* cbT;     // (6,32,1024) transposed normalized codebook
  float* cbn2;    // (6,1024) ||cb||^2
  float* loss;    // 1 accumulator
  float* wdp;     // (6,864,32)  zero-padded Wd
  float* wup;     // (6,32,864)  zero-padded Wu
  float* bup;     // (6,864)     zero-padded bu
};

__device__ __forceinline__ void amin2(float& bs, int& bc, float os, int oc) {
  if (os < bs || (os == bs && oc < bc)) { bs = os; bc = oc; }
}

// ---- Kernel 1a: normalize codebooks into workspace, zero loss accumulator ----
__global__ void prep_codebook(Params p) {
  if (blockIdx.x == 0 && threadIdx.x == 0) p.loss[0] = 0.f;
  int lane = threadIdx.x & 31;
  int wid  = threadIdx.x >> 5;
  int row  = blockIdx.x * NW + wid;           // 0 .. 6*1024-1
  if (row >= NVQ * KCODE) return;
  int g = row >> 10, k = row & (KCODE - 1);
  float e = p.emb[g][k * CDIM + lane];
  float sq = e * e;
  for (int off = 16; off > 0; off >>= 1) sq += __shfl_xor(sq, off, 32);
  float cl = fmaxf(sqrtf(sq), 1e-12f);
  float v = e / cl;
  p.cbrow[((size_t)g * KCODE + k) * CDIM + lane] = v;
  p.cbT[(size_t)g * CDIM * KCODE + (size_t)lane * KCODE + k] = v;
  if (lane == 0) p.cbn2[g * KCODE + k] = sq / (cl * cl);
}

// ---- Kernel 1b: zero-pad the ragged weight matrices (kills all inner-loop guards)
__global__ void prep_weights(Params p) {
  const int WDN = NVQ * KPAD * CDIM;          // 165888
  int idx = blockIdx.x * 256 + threadIdx.x;
  if (idx < WDN) {                            // Wd_pad[g][k][c]
    int g = idx / (KPAD * CDIM), r = idx % (KPAD * CDIM);
    int k = r / CDIM, c = r % CDIM;
    int d = (g == NVQ - 1) ? 855 : 853;
    p.wdp[idx] = (k < d) ? p.Wd[g][k * CDIM + c] : 0.f;
    return;
  }
  idx -= WDN;
  if (idx < WDN) {                            // Wu_pad[g][k][n]
    int g = idx / (CDIM * KPAD), r = idx % (CDIM * KPAD);
    int k = r / KPAD, n = r % KPAD;
    int d = (g == NVQ - 1) ? 855 : 853;
    p.wup[WDN + -WDN + idx + 0] = 0.f;  // placeholder overwritten below
    p.wup[idx] = (n < d) ? p.Wu[g][k * d + n] : 0.f;
    return;
  }
  idx -= WDN;
  if (idx < NVQ * KPAD) {                     // bu_pad[g][n]
    int g = idx / KPAD, n = idx % KPAD;
    int d = (g == NVQ - 1) ? 855 : 853;
    p.bup[idx] = (n < d) ? p.bu[g][n] : 0.f;
  }
}

// ---- Kernel 2: main fused PVQ forward, one 16-row tile per block ----
__global__ __launch_bounds__(256) void pvq_main(Params p) {
  __shared__ float lds_tile[16 * TSTR];     // zi tile, reused as out tile
  __shared__ float lds_zdpre[16 * CDIM];
  __shared__ float lds_zd[16 * 33];         // padded stride 33 (odd -> no conflicts)
  __shared__ float lds_zq[16 * 33];
  __shared__ float lds_bestv[NW * 16];
  __shared__ int   lds_bestc[NW * 16];
  __shared__ int   lds_code[16];
  __shared__ float lds_norm[16];
  __shared__ float lds_ls[NW];

  const int tid  = threadIdx.x;
  const int lane = tid & 31;
  const int w    = tid >> 5;
  const int half = lane >> 4;     // 0: K+0/1 | 1: K+2/3 (per 16x16x4 f32 A/B layout)
  const int hl   = lane & 15;
  const int row0 = blockIdx.x * 16;

  for (int g = 0; g < NVQ; ++g) {
    const int s = 853 * g;
    const int d = (g == NVQ - 1) ? 855 : 853;
    const float* bd  = p.bd[g];
    const float* wdp = p.wdp + (size_t)g * KPAD * CDIM;
    const float* wup = p.wup + (size_t)g * CDIM * KPAD;
    const float* bup = p.bup + (size_t)g * KPAD;

    // ---- Phase A: init zd_pre with bias, zero K pad, gather zi tile (coalesced runs)
    for (int i = tid; i < 16 * CDIM; i += 256) lds_zdpre[i] = bd[i & 31];
    {
      int padn = KPAD - d;                   // 11 or 9
      for (int i = tid; i < 16 * padn; i += 256)
        lds_tile[(i / padn) * TSTR + d + (i % padn)] = 0.f;
    }
    for (int rr = 0; rr < 2; ++rr) {
      int r = w * 2 + rr;
      int row = row0 + r;
      int b = row / TOV, t = row % TOV;
      const float* zb = p.z_e + (size_t)b * (HH * WT) * CC;
      for (int ho = 0; ho < HH * OVV; ++ho) {
        int h = ho / OVV, o = ho % OVV;
        int base = o * FIXD + h;
        int lo = s - base;
        int clo = lo <= 0 ? 0 : (lo + 4) / 5;
        int hi = s + d - base;
        int chi = hi <= 0 ? 0 : (hi + 4) / 5;
        if (chi > CC) chi = CC;
        const float* src = zb + (size_t)(h * WT + t * OVV + o) * CC;
        for (int c = clo + lane; c < chi; c += 32)
          lds_tile[r * TSTR + (base + 5 * c - s)] = src[c];
      }
    }
    __syncthreads();

    // ---- Phase B: down-projection zd_pre = zi @ Wd (+bd), K split across 8 waves
    {
      v8f acc0 = {0,0,0,0,0,0,0,0}, acc1 = {0,0,0,0,0,0,0,0};
      int kbase = w * (KPAD / NW);           // 108 per wave
      for (int i = 0; i < (KPAD / NW) / 4; ++i) {  // 27 WMMA k-steps
        int kk = kbase + 4 * i + half * 2;
        v2f a; a.x = lds_tile[hl * TSTR + kk];
               a.y = lds_tile[hl * TSTR + kk + 1];
        v2f b0, b1;
        b0.x = wdp[(kk    ) * CDIM + hl];
        b0.y = wdp[(kk + 1) * CDIM + hl];
        b1.x = wdp[(kk    ) * CDIM + 16 + hl];
        b1.y = wdp[(kk + 1) * CDIM + 16 + hl];
        acc0 = __builtin_amdgcn_wmma_f32_16x16x4_f32(false, a, false, b0, (short)0, acc0, false, false);
        acc1 = __builtin_amdgcn_wmma_f32_16x16x4_f32(false, a, false, b1, (short)0, acc1, false, false);
      }
      for (int v = 0; v < 8; ++v) {
        int r = v + half * 8;
        atomicAdd(&lds_zdpre[r * CDIM + hl],      acc0[v]);
        atomicAdd(&lds_zdpre[r * CDIM + 16 + hl], acc1[v]);
      }
    }
    __syncthreads();

    // ---- Phase C: L2 normalize rows; write z_e_downs
    if (tid < 16) {
      float ss = 0.f;
      for (int c = 0; c < CDIM; ++c) { float x = lds_zdpre[tid * CDIM + c]; ss += x * x; }
      lds_norm[tid] = fmaxf(sqrtf(ss), 1e-12f);
    }
    __syncthreads();
    for (int i = tid; i < 16 * CDIM; i += 256) {
      int r = i >> 5, c = i & 31;
      float zdv = lds_zdpre[i] / lds_norm[r];
      lds_zd[r * 33 + c] = zdv;
      int grow = row0 + r; int b = grow / TOV, t = grow % TOV;
      p.out[ZD_OFF + ((((size_t)b * NVQ + g) * TOV + t) * CDIM) + c] = zdv;
    }
    __syncthreads();

    // ---- Phase D: scores = (-2*zd) @ cbT + ||cb||^2 ; argmin over 1024 codes
    {
      float bestS[8]; int bestC[8];
      for (int v = 0; v < 8; ++v) { bestS[v] = 3.0e38f; bestC[v] = 0; }
      const float* cbT = p.cbT + (size_t)g * CDIM * KCODE;
      for (int ti = 0; ti < 8; ++ti) {
        int n0 = (w * 8 + ti) * 16;
        float ci = p.cbn2[g * KCODE + n0 + hl];
        v8f acc = {ci, ci, ci, ci, ci, ci, ci, ci};
        for (int i = 0; i < CDIM / 4; ++i) {
          int kk = 4 * i + half * 2;
          v2f a; a.x = -2.f * lds_zd[hl * 33 + kk];
                 a.y = -2.f * lds_zd[hl * 33 + kk + 1];
          v2f b; b.x = cbT[(size_t)(kk    ) * KCODE + n0 + hl];
                 b.y = cbT[(size_t)(kk + 1) * KCODE + n0 + hl];
          acc = __builtin_amdgcn_wmma_f32_16x16x4_f32(false, a, false, b, (short)0, acc, false, false);
        }
        int col = n0 + hl;
        for (int v = 0; v < 8; ++v) amin2(bestS[v], bestC[v], acc[v], col);
      }
      for (int off = 8; off >= 1; off >>= 1)
        for (int v = 0; v < 8; ++v) {
          float os = __shfl_xor(bestS[v], off, 32);
          int   oc = __shfl_xor(bestC[v], off, 32);
          amin2(bestS[v], bestC[v], os, oc);
        }
      if (hl == 0)
        for (int v = 0; v < 8; ++v) {
          int r = v + half * 8;
          lds_bestv[w * 16 + r] = bestS[v];
          lds_bestc[w * 16 + r] = bestC[v];
        }
    }
    __syncthreads();
    if (tid < 16) {
      float bs = lds_bestv[tid]; int bc = lds_bestc[tid];
      for (int ww = 1; ww < NW; ++ww)
        amin2(bs, bc, lds_bestv[ww * 16 + tid], lds_bestc[ww * 16 + tid]);
      lds_code[tid] = bc;
      int grow = row0 + tid; int b = grow / TOV, t = grow % TOV;
      p.out[IDX_OFF + ((size_t)b * NVQ + g) * TOV + t] = (float)bc;
    }
    __syncthreads();

    // ---- Phase E: gather zq = cb[code]; accumulate loss sum((zd - zq)^2)
    {
      float lsum = 0.f;
      for (int i = tid; i < 16 * CDIM; i += 256) {
        int r = i >> 5, c = i & 31;
        float qv = p.cbrow[(((size_t)g * KCODE + lds_code[r]) * CDIM) + c];
        lds_zq[r * 33 + c] = qv;
        float dlt = lds_zd[r * 33 + c] - qv;
        lsum += dlt * dlt;
      }
      for (int off = 16; off > 0; off >>= 1) lsum += __shfl_xor(lsum, off, 32);
      if (lane == 0) lds_ls[w] = lsum;
    }
    __syncthreads();
    if (tid == 0) {
      float t2 = 0.f;
      for (int ww = 0; ww < NW; ++ww) t2 += lds_ls[ww];
      atomicAdd(p.loss, t2);
    }
    __syncthreads();

    // ---- Phase F: up-projection out = zq @ Wu + bu, staged into lds_tile
    for (int nt = w; nt < (d + 15) / 16; nt += NW) {
      int n0 = nt * 16;
      int col = n0 + hl;
      float ci = bup[col];
      v8f acc = {ci, ci, ci, ci, ci, ci, ci, ci};
      for (int i = 0; i < CDIM / 4; ++i) {
        int kk = 4 * i + half * 2;
        v2f a; a.x = lds_zq[hl * 33 + kk];
               a.y = lds_zq[hl * 33 + kk + 1];
        v2f b; b.x = wup[(size_t)(kk    ) * KPAD + col];
               b.y = wup[(size_t)(kk + 1) * KPAD + col];
        acc = __builtin_amdgcn_wmma_f32_16x16x4_f32(false, a, false, b, (short)0, acc, false, false);
      }
      if (col < d)
        for (int v = 0; v < 8; ++v)
          lds_tile[(v + half * 8) * TSTR + col] = acc[v];
    }
    __syncthreads();

    // ---- Phase G: scatter out tile to zq output via coalesced runs
    for (int rr = 0; rr < 2; ++rr) {
      int r = w * 2 + rr;
      int row = row0 + r;
      int b = row / TOV, t = row % TOV;
      float* db = p.out + (size_t)b * (HH * WT) * CC;
      for (int ho = 0; ho < HH * OVV; ++ho) {
        int h = ho / OVV, o = ho % OVV;
        int base = o * FIXD + h;
        int lo = s - base;
        int clo = lo <= 0 ? 0 : (lo + 4) / 5;
        int hi = s + d - base;
        int chi = hi <= 0 ? 0 : (hi + 4) / 5;
        if (chi > CC) chi = CC;
        float* dst = db + (size_t)(h * WT + t * OVV + o) * CC;
        for (int c = clo + lane; c < chi; c += 32)
          dst[c] = lds_tile[r * TSTR + (base + 5 * c - s)];
      }
    }
    __syncthreads();
  }
}

// ---- Kernel 3: finalize the two (numerically equal) loss scalars ----
__global__ void finalize_loss(Params p) {
  float v = p.loss[0] / ((float)(NROWS * CDIM)) / (float)NVQ;
  p.out[CM_OFF]     = v;
  p.out[CM_OFF + 1] = v;
}

extern "C" void kernel_launch(void* const* d_in, const int* in_sizes, int n_in,
                              void* d_out, int out_size, void* d_ws, size_t ws_size,
                              hipStream_t stream) {
  (void)in_sizes; (void)n_in; (void)out_size; (void)ws_size;
  Params p;
  p.z_e = (const float*)d_in[0];
  for (int g = 0; g < NVQ; ++g) {
    p.Wd[g]  = (const float*)d_in[1 + 5 * g + 0];
    p.bd[g]  = (const float*)d_in[1 + 5 * g + 1];
    p.Wu[g]  = (const float*)d_in[1 + 5 * g + 2];
    p.bu[g]  = (const float*)d_in[1 + 5 * g + 3];
    p.emb[g] = (const float*)d_in[1 + 5 * g + 4];
  }
  p.out = (float*)d_out;
  float* wsf = (float*)d_ws;
  p.cbrow = wsf;                 // 6*1024*32 = 196608 floats
  p.cbT   = wsf + 196608;        // 196608 floats
  p.cbn2  = wsf + 393216;        // 6144 floats
  p.loss  = wsf + 399360;        // 1 float (+63 pad)
  p.wdp   = wsf + 399424;        // 6*864*32 = 165888 floats
  p.wup   = wsf + 565312;        // 6*32*864 = 165888 floats
  p.bup   = wsf + 731200;        // 6*864    = 5184 floats
  prep_codebook<<<(NVQ * KCODE) / NW, 256, 0, stream>>>(p);
  {
    int total = 2 * (NVQ * KPAD * CDIM) + NVQ * KPAD;  // 336960
    prep_weights<<<(total + 255) / 256, 256, 0, stream>>>(p);
  }
  pvq_main<<<NROWS / 16, 256, 0, stream>>>(p);
  finalize_loss<<<1, 1, 0, stream>>>(p);
}